// EmbeddingBlock_13005160972691
// MI455X (gfx1250) — compile-verified
//
#include <hip/hip_runtime.h>

typedef __attribute__((ext_vector_type(2))) float v2f;
typedef __attribute__((ext_vector_type(4))) float v4f;
typedef __attribute__((ext_vector_type(8))) float v8f;
typedef __attribute__((ext_vector_type(2))) int   v2i;

#define EMBED_DIM 128
#define VOCAB 100
#define N_RBF 6

// ---------------------------------------------------------------------------
// Kernel 1: precompute folded tables with V_WMMA_F32_16X16X4_F32.
//   P1 = emb_table @ W_dense[  0:128]   (100x128)
//   P2 = emb_table @ W_dense[128:256]   (100x128)
//   Wc = W_edge    @ W_dense[256:384]   (  6x128)
// grid = (8 ntiles, 8 mtiles): mtile 0..6 -> P1/P2 rows, mtile 7 -> Wc.
// One wave (32 lanes) per 16x16 output tile; K loop over 128 in steps of 4.
// Row padding is done branchlessly (clamped index + multiplicative mask) so
// the inner loop has no EXEC-divergent loads and WMMA sees EXEC == all-1s.
// ---------------------------------------------------------------------------
__global__ __launch_bounds__(32)
void precompute_tables(const float* __restrict__ emb,
                       const float* __restrict__ W_edge,
                       const float* __restrict__ W_dense,
                       float* __restrict__ P1,
                       float* __restrict__ P2,
                       float* __restrict__ Wc) {
  const int lane  = threadIdx.x;       // 0..31
  const int ntile = blockIdx.x;        // 0..7  (N direction)
  const int mtile = blockIdx.y;        // 0..7  (7 == Wc tile)
  const int half  = lane >> 4;         // lane group 0 / 1
  const int lm    = lane & 15;
  const int n     = ntile * 16 + lm;   // B/C/D column owned by this lane

  const float* __restrict__ Wd1 = W_dense;                        // rows 0..127
  const float* __restrict__ Wd2 = W_dense + 128 * EMBED_DIM;      // rows 128..255
  const float* __restrict__ Wd3 = W_dense + 256 * EMBED_DIM;      // rows 256..383

  v8f acc0 = {};
  v8f acc1 = {};

  if (mtile < 7) {
    const int   m     = mtile * 16 + lm;            // A row owned by this lane
    const int   msafe = (m < VOCAB) ? m : 0;        // clamped (branchless pad)
    const float mmask = (m < VOCAB) ? 1.0f : 0.0f;
    const float* __restrict__ arow = emb + msafe * EMBED_DIM;
    for (int k0 = 0; k0 < EMBED_DIM; k0 += 4) {
      const int ka = k0 + 2 * half;                 // K pair for this lane half
      v2f a;
      a.x = mmask * arow[ka + 0];
      a.y = mmask * arow[ka + 1];
      v2f b1, b2;
      b1.x = Wd1[(ka + 0) * EMBED_DIM + n];
      b1.y = Wd1[(ka + 1) * EMBED_DIM + n];
      b2.x = Wd2[(ka + 0) * EMBED_DIM + n];
      b2.y = Wd2[(ka + 1) * EMBED_DIM + n];
      acc0 = __builtin_amdgcn_wmma_f32_16x16x4_f32(false, a, false, b1,
                                                   (short)0, acc0, false, false);
      acc1 = __builtin_amdgcn_wmma_f32_16x16x4_f32(false, a, false, b2,
                                                   (short)0, acc1, false, false);
    }
#pragma unroll
    for (int v = 0; v < 8; ++v) {
      const int m_out = mtile * 16 + v + 8 * half;   // C/D VGPR layout
      if (m_out < VOCAB) {
        P1[m_out * EMBED_DIM + n] = acc0[v];
        P2[m_out * EMBED_DIM + n] = acc1[v];
      }
    }
  } else {
    const int   msafe = (lm < N_RBF) ? lm : 0;
    const float mmask = (lm < N_RBF) ? 1.0f : 0.0f;
    const float* __restrict__ arow = W_edge + msafe * EMBED_DIM;
    for (int k0 = 0; k0 < EMBED_DIM; k0 += 4) {
      const int ka = k0 + 2 * half;
      v2f a;
      a.x = mmask * arow[ka + 0];
      a.y = mmask * arow[ka + 1];
      v2f b3;
      b3.x = Wd3[(ka + 0) * EMBED_DIM + n];
      b3.y = Wd3[(ka + 1) * EMBED_DIM + n];
      acc0 = __builtin_amdgcn_wmma_f32_16x16x4_f32(false, a, false, b3,
                                                   (short)0, acc0, false, false);
    }
#pragma unroll
    for (int v = 0; v < 8; ++v) {
      const int m_out = v + 8 * half;
      if (m_out < N_RBF) Wc[m_out * EMBED_DIM + n] = acc0[v];
    }
  }
}

// ---------------------------------------------------------------------------
// Kernel 2: streaming edge kernel (bandwidth-bound, ~330 MB total traffic).
//   out[e] = silu( P1[z[nbr0]] + P2[z[nbr1]] + e_rbf[e] . Wc + b )
// One wave per edge, lane owns 4 contiguous floats (b128 load/store path).
// Wc + b cached in LDS; P1/P2 (51 KB each) stay hot in L2.
// nbr row loaded as one b64 (int2), rbf row as three b64 (v2f).
// Output is streamed once -> non-temporal b128 stores.
// ---------------------------------------------------------------------------
__global__ __launch_bounds__(256)
void edge_fused(const float* __restrict__ e_rbf,
                const int*   __restrict__ z,
                const int*   __restrict__ nbr,
                const float* __restrict__ P1,
                const float* __restrict__ P2,
                const float* __restrict__ Wc,
                const float* __restrict__ b_dense,
                float* __restrict__ out,
                int E) {
  __shared__ float sWc[N_RBF * EMBED_DIM];
  __shared__ float sB[EMBED_DIM];
  for (int i = threadIdx.x; i < N_RBF * EMBED_DIM; i += blockDim.x) sWc[i] = Wc[i];
  if (threadIdx.x < EMBED_DIM) sB[threadIdx.x] = b_dense[threadIdx.x];
  __syncthreads();

  const int lane  = threadIdx.x & 31;
  const int wid   = threadIdx.x >> 5;
  const int wpb   = blockDim.x >> 5;
  const int gwave = blockIdx.x * wpb + wid;
  const int nwave = gridDim.x * wpb;
  const int d     = lane * 4;          // 4 outputs per lane, 128 per wave

  const v4f bb = *(const v4f*)(sB + d);
  v4f w[N_RBF];
#pragma unroll
  for (int k = 0; k < N_RBF; ++k) w[k] = *(const v4f*)(sWc + k * EMBED_DIM + d);

  for (int e = gwave; e < E; e += nwave) {
    const v2i nn = *(const v2i*)(nbr + 2 * e);         // one b64
    const int i0 = z[nn.x];
    const int i1 = z[nn.y];

    const float* __restrict__ rb = e_rbf + e * N_RBF;  // three b64
    const v2f r01 = *(const v2f*)(rb + 0);
    const v2f r23 = *(const v2f*)(rb + 2);
    const v2f r45 = *(const v2f*)(rb + 4);
    const float r[N_RBF] = {r01.x, r01.y, r23.x, r23.y, r45.x, r45.y};

    const v4f p = *(const v4f*)(P1 + i0 * EMBED_DIM + d);  // b128 gathers
    const v4f q = *(const v4f*)(P2 + i1 * EMBED_DIM + d);

    v4f a;
    a.x = p.x + q.x + bb.x;
    a.y = p.y + q.y + bb.y;
    a.z = p.z + q.z + bb.z;
    a.w = p.w + q.w + bb.w;
#pragma unroll
    for (int k = 0; k < N_RBF; ++k) {
      a.x = fmaf(r[k], w[k].x, a.x);
      a.y = fmaf(r[k], w[k].y, a.y);
      a.z = fmaf(r[k], w[k].z, a.z);
      a.w = fmaf(r[k], w[k].w, a.w);
    }
    // silu(x) = x * sigmoid(x)
    a.x = a.x / (1.0f + __expf(-a.x));
    a.y = a.y / (1.0f + __expf(-a.y));
    a.z = a.z / (1.0f + __expf(-a.z));
    a.w = a.w / (1.0f + __expf(-a.w));

    __builtin_nontemporal_store(a, (v4f*)(out + (size_t)e * EMBED_DIM + d));
  }
}

extern "C" void kernel_launch(void* const* d_in, const int* in_sizes, int n_in,
                              void* d_out, int out_size, void* d_ws, size_t ws_size,
                              hipStream_t stream) {
  const float* e_rbf   = (const float*)d_in[0];
  const int*   z       = (const int*)d_in[1];
  const int*   nbr     = (const int*)d_in[2];
  const float* W_edge  = (const float*)d_in[3];
  const float* emb     = (const float*)d_in[4];
  const float* W_dense = (const float*)d_in[5];
  const float* b_dense = (const float*)d_in[6];
  float*       out     = (float*)d_out;

  float* ws = (float*)d_ws;
  float* P1 = ws;                         // 100*128 floats
  float* P2 = ws + VOCAB * EMBED_DIM;     // 100*128 floats
  float* Wc = ws + 2 * VOCAB * EMBED_DIM; // 6*128 floats

  const int E = in_sizes[0] / N_RBF;      // 600000

  hipLaunchKernelGGL(precompute_tables, dim3(8, 8), dim3(32), 0, stream,
                     emb, W_edge, W_dense, P1, P2, Wc);
  hipLaunchKernelGGL(edge_fused, dim3(4096), dim3(256), 0, stream,
                     e_rbf, z, nbr, P1, P2, Wc, b_dense, out, E);
}